// RobertaGraphEncoder_12919261626719
// MI455X (gfx1250) — compile-verified
//
#include <hip/hip_runtime.h>
#include <hip/hip_bf16.h>

typedef float v2f __attribute__((ext_vector_type(2)));
typedef float v8f __attribute__((ext_vector_type(8)));

#define FEAT_IN 768
#define WORD_IN 300
#define HIDDEN  64

// ---------------------------------------------------------------------------
// fp32 WMMA GEMM: C[M,N] = A[M,K] @ B[K,N] (+ bias[n]).
// One wave computes a 16x64 strip of C: 4 accumulators, so each A fragment
// (16x4 f32, 2 VGPRs) feeds 4 independent V_WMMA_F32_16X16X4_F32 issues.
// Requires M%16==0, N%64==0, K%4==0.
// ---------------------------------------------------------------------------
__global__ void k_gemm_f32_wmma(const float* __restrict__ A,
                                const float* __restrict__ B,
                                const float* __restrict__ bias,
                                float* __restrict__ C,
                                int Mt, int Nb, int K,
                                int lda, int ldb, int ldc) {
    int wave = blockIdx.x * (blockDim.x >> 5) + (threadIdx.x >> 5);
    if (wave >= Mt * Nb) return;
    int tm = wave / Nb;
    int nb = wave - tm * Nb;          // 64-wide column block
    int lane = threadIdx.x & 31;
    int half = lane >> 4;             // 0: K pair {k,k+1}, 1: {k+2,k+3}
    int l16  = lane & 15;

    // A fragment source: lane l16 owns row m0+l16 (ISA 16x4 f32 A layout).
    const float* __restrict__ Ap = A + (size_t)(tm * 16 + l16) * lda + half * 2;
    // B fragment source: lane l16 owns col n0 + t*16 + l16.
    const float* __restrict__ Bp = B + (size_t)(half * 2) * ldb + nb * 64 + l16;

    v8f acc0 = {}, acc1 = {}, acc2 = {}, acc3 = {};
    for (int k = 0; k < K; k += 4) {
        v2f a;
        a.x = Ap[k];
        a.y = Ap[k + 1];
        const float* __restrict__ brow0 = Bp + (size_t)k * ldb;
        const float* __restrict__ brow1 = Bp + (size_t)(k + 1) * ldb;
        v2f b0, b1, b2, b3;
        b0.x = brow0[0];   b0.y = brow1[0];
        b1.x = brow0[16];  b1.y = brow1[16];
        b2.x = brow0[32];  b2.y = brow1[32];
        b3.x = brow0[48];  b3.y = brow1[48];
        acc0 = __builtin_amdgcn_wmma_f32_16x16x4_f32(false, a, false, b0, (short)0, acc0, false, false);
        acc1 = __builtin_amdgcn_wmma_f32_16x16x4_f32(false, a, false, b1, (short)0, acc1, false, false);
        acc2 = __builtin_amdgcn_wmma_f32_16x16x4_f32(false, a, false, b2, (short)0, acc2, false, false);
        acc3 = __builtin_amdgcn_wmma_f32_16x16x4_f32(false, a, false, b3, (short)0, acc3, false, false);
    }

    // C layout per tile: VGPR r -> row (half*8 + r), col l16.
    float* __restrict__ Cp = C + (size_t)(tm * 16 + half * 8) * ldc + nb * 64 + l16;
    float bn0 = 0.0f, bn1 = 0.0f, bn2 = 0.0f, bn3 = 0.0f;
    if (bias) {
        const float* __restrict__ bp = bias + nb * 64 + l16;
        bn0 = bp[0]; bn1 = bp[16]; bn2 = bp[32]; bn3 = bp[48];
    }
    #pragma unroll
    for (int r = 0; r < 8; ++r) {
        float* __restrict__ crow = Cp + (size_t)r * ldc;
        crow[0]  = acc0[r] + bn0;
        crow[16] = acc1[r] + bn1;
        crow[32] = acc2[r] + bn2;
        crow[48] = acc3[r] + bn3;
    }
}

// x[0 : n_doc*768] = doc_features (float4-vectorized; count % 4 == 0)
__global__ void k_copy_doc(const float4* __restrict__ src, float4* __restrict__ dst,
                           long long count4) {
    long long i = (long long)blockIdx.x * blockDim.x + threadIdx.x;
    if (i < count4) dst[i] = src[i];
}

// deg[n] = 1.0 (self loop weight)
__global__ void k_deg_init(float* __restrict__ deg, int n) {
    int i = blockIdx.x * blockDim.x + threadIdx.x;
    if (i < n) deg[i] = 1.0f;
}

// deg[col[e]] += w[e]
__global__ void k_deg_acc(const int* __restrict__ ei, const float* __restrict__ ew,
                          float* __restrict__ deg, int E) {
    int e = blockIdx.x * blockDim.x + threadIdx.x;
    if (e >= E) return;
    int c = ei[E + e];
    __hip_atomic_fetch_add(&deg[c], ew[e], __ATOMIC_RELAXED, __HIP_MEMORY_SCOPE_AGENT);
}

// dis[n] = rsqrt(max(deg,1e-12))
__global__ void k_dis(const float* __restrict__ deg, float* __restrict__ dis, int n) {
    int i = blockIdx.x * blockDim.x + threadIdx.x;
    if (i >= n) return;
    float d = deg[i];
    dis[i] = d > 0.0f ? rsqrtf(fmaxf(d, 1e-12f)) : 0.0f;
}

// out[n, f] = bias[f] + xw[n, f] * dis[n]^2   (bias + self-loop edge fused)
__global__ void k_out_init(const float* __restrict__ xw, const float* __restrict__ dis,
                           const float* __restrict__ bias, float* __restrict__ out,
                           int n_nodes) {
    long long i = (long long)blockIdx.x * blockDim.x + threadIdx.x;
    if (i >= (long long)n_nodes * HIDDEN) return;
    int n = (int)(i >> 6);
    int f = (int)(i & 63);
    float d = dis[n];
    out[i] = bias[f] + xw[i] * d * d;
}

// One wave per edge: out[col] += xw[row] * (dis[row]*w*dis[col]) over 64 feats.
__global__ void k_scatter(const int* __restrict__ ei, const float* __restrict__ ew,
                          const float* __restrict__ dis, const float* __restrict__ xw,
                          float* __restrict__ out, int E) {
    int e = blockIdx.x * (blockDim.x >> 5) + (threadIdx.x >> 5);
    if (e >= E) return;
    int lane = threadIdx.x & 31;
    int r = ei[e];
    int c = ei[E + e];
    float coef = dis[r] * ew[e] * dis[c];
    const float2* __restrict__ src = (const float2*)(xw + (size_t)r * HIDDEN);
    float2 v = src[lane];
    float* __restrict__ dst = out + (size_t)c * HIDDEN + lane * 2;
    __hip_atomic_fetch_add(dst,     v.x * coef, __ATOMIC_RELAXED, __HIP_MEMORY_SCOPE_AGENT);
    __hip_atomic_fetch_add(dst + 1, v.y * coef, __ATOMIC_RELAXED, __HIP_MEMORY_SCOPE_AGENT);
}

__global__ void k_relu(float* __restrict__ h, long long count) {
    long long i = (long long)blockIdx.x * blockDim.x + threadIdx.x;
    if (i < count) h[i] = fmaxf(h[i], 0.0f);
}

// out[i, f] = h[mask[i], f];  out[M*64 + i] = (float)y[mask[i]]
__global__ void k_gather(const float* __restrict__ h, const int* __restrict__ mask,
                         const int* __restrict__ y, float* __restrict__ out, int M) {
    int i = blockIdx.x * blockDim.x + threadIdx.x;
    if (i >= M * HIDDEN) return;
    int r = i >> 6;
    int f = i & 63;
    int n = mask[r];
    out[i] = h[(size_t)n * HIDDEN + f];
    if (f == 0) out[(size_t)M * HIDDEN + r] = (float)y[n];
}

extern "C" void kernel_launch(void* const* d_in, const int* in_sizes, int n_in,
                              void* d_out, int out_size, void* d_ws, size_t ws_size,
                              hipStream_t stream) {
    const float* doc    = (const float*)d_in[0];
    const float* wordf  = (const float*)d_in[1];
    const float* ew     = (const float*)d_in[2];
    const float* W_lin  = (const float*)d_in[3];
    const float* b_lin  = (const float*)d_in[4];
    const float* W1     = (const float*)d_in[5];
    const float* b1     = (const float*)d_in[6];
    const float* W2     = (const float*)d_in[7];
    const float* b2     = (const float*)d_in[8];
    const int*   ei     = (const int*)d_in[9];
    const int*   mask   = (const int*)d_in[10];
    const int*   y      = (const int*)d_in[11];

    const int n_doc   = in_sizes[0] / FEAT_IN;  // 20000
    const int n_word  = in_sizes[1] / WORD_IN;  // 80000
    const int n_nodes = n_doc + n_word;         // 100000
    const int E = in_sizes[2];                  // 1,000,000
    const int M = in_sizes[10];                 // 10000

    // workspace carve-up (floats)
    float* x   = (float*)d_ws;                          // n_nodes * 768
    float* deg = x   + (size_t)n_nodes * FEAT_IN;       // n_nodes
    float* dis = deg + n_nodes;                         // n_nodes
    float* xw  = dis + n_nodes;                         // n_nodes * 64 (reused)
    float* h1  = xw  + (size_t)n_nodes * HIDDEN;        // n_nodes * 64
    float* h2  = h1  + (size_t)n_nodes * HIDDEN;        // n_nodes * 64

    const int TPB = 256;            // 8 waves/block
    const int WPB = TPB / 32;

    // 1) x[0:n_doc] = doc_features
    {
        long long cnt4 = (long long)n_doc * FEAT_IN / 4;
        int blocks = (int)((cnt4 + TPB - 1) / TPB);
        k_copy_doc<<<blocks, TPB, 0, stream>>>((const float4*)doc, (float4*)x, cnt4);
    }
    // 2) x[n_doc:] = word_features @ W_lin + b_lin  (80000 x 300 x 768)
    {
        int Mt = n_word / 16, Nb = FEAT_IN / 64;
        int blocks = (Mt * Nb + WPB - 1) / WPB;
        k_gemm_f32_wmma<<<blocks, TPB, 0, stream>>>(
            wordf, W_lin, b_lin, x + (size_t)n_doc * FEAT_IN,
            Mt, Nb, WORD_IN, WORD_IN, FEAT_IN, FEAT_IN);
    }
    // 3) degree + rsqrt norm
    k_deg_init<<<(n_nodes + TPB - 1) / TPB, TPB, 0, stream>>>(deg, n_nodes);
    k_deg_acc<<<(E + TPB - 1) / TPB, TPB, 0, stream>>>(ei, ew, deg, E);
    k_dis<<<(n_nodes + TPB - 1) / TPB, TPB, 0, stream>>>(deg, dis, n_nodes);

    // 4) layer 1: xw = x @ W1 (100000 x 768 x 64)
    {
        int Mt = n_nodes / 16, Nb = HIDDEN / 64;
        int blocks = (Mt * Nb + WPB - 1) / WPB;
        k_gemm_f32_wmma<<<blocks, TPB, 0, stream>>>(
            x, W1, nullptr, xw, Mt, Nb, FEAT_IN, FEAT_IN, HIDDEN, HIDDEN);
    }
    // 5) h1 = b1 + self-loop, edge scatter, ReLU
    {
        long long cnt = (long long)n_nodes * HIDDEN;
        int blocks = (int)((cnt + TPB - 1) / TPB);
        k_out_init<<<blocks, TPB, 0, stream>>>(xw, dis, b1, h1, n_nodes);
        k_scatter<<<(E + WPB - 1) / WPB, TPB, 0, stream>>>(ei, ew, dis, xw, h1, E);
        k_relu<<<blocks, TPB, 0, stream>>>(h1, cnt);
    }
    // 6) layer 2: xw = h1 @ W2 (100000 x 64 x 64)
    {
        int Mt = n_nodes / 16, Nb = HIDDEN / 64;
        int blocks = (Mt * Nb + WPB - 1) / WPB;
        k_gemm_f32_wmma<<<blocks, TPB, 0, stream>>>(
            h1, W2, nullptr, xw, Mt, Nb, HIDDEN, HIDDEN, HIDDEN, HIDDEN);
    }
    // 7) h2 = b2 + self-loop, edge scatter
    {
        long long cnt = (long long)n_nodes * HIDDEN;
        int blocks = (int)((cnt + TPB - 1) / TPB);
        k_out_init<<<blocks, TPB, 0, stream>>>(xw, dis, b2, h2, n_nodes);
        k_scatter<<<(E + WPB - 1) / WPB, TPB, 0, stream>>>(ei, ew, dis, xw, h2, E);
    }
    // 8) masked gather of features + labels
    k_gather<<<(M * HIDDEN + TPB - 1) / TPB, TPB, 0, stream>>>(h2, mask, y, (float*)d_out, M);
}